// SpectralConv2d_10196252360930
// MI455X (gfx1250) — compile-verified
//
#include <hip/hip_runtime.h>

// ---------------------------------------------------------------------------
// SpectralConv2d on MI455X (gfx1250).
//   dht2d(X) = T * X * T^T where T is the 128x128 linear map of the
//   reference's recursive butterfly (DIT Cooley-Tukey with cas twiddles).
//   Only the two 32x32 mode corners feed the output, so we compute a 64x32
//   slice per image (WMMA fp32 GEMMs), do the bandwidth-bound mode mixing
//   with a coalesced kernel using double-buffered GLOBAL_LOAD_ASYNC_TO_LDS,
//   and invert with two more WMMA fp32 GEMMs.
// ---------------------------------------------------------------------------

typedef float v2f __attribute__((ext_vector_type(2)));
typedef float v8f __attribute__((ext_vector_type(8)));
typedef int   i32x4 __attribute__((vector_size(16)));   // matches builtin sig
typedef __attribute__((address_space(1))) i32x4 gas_i32x4;
typedef __attribute__((address_space(3))) i32x4 las_i32x4;

#define B_SZ   8
#define CIN    256
#define COUT   256
#define MDIM   128
#define NDIM   128

#if __has_builtin(__builtin_amdgcn_global_load_async_to_lds_b128)
#define HAS_ASYNC_LDS 1
#else
#define HAS_ASYNC_LDS 0
#endif

// row selector for the two corner row-bands {0..31} U {96..127}
__device__ __forceinline__ int rsel(int r) { return (r < 32) ? r : r + 64; }

// ---------------------------------------------------------------------------
// Kernel 0: build T (128x128) by pushing the 128 unit vectors through the
// iterative butterfly in LDS. T[n][c] = butterfly(e_c)[n].
// ---------------------------------------------------------------------------
__global__ void build_T(float* __restrict__ T) {
  __shared__ float S[128 * 128];
  const int t = threadIdx.x;  // 256 threads
  for (int idx = t; idx < 128 * 128; idx += 256) {
    int n = idx >> 7, c = idx & 127;
    int rc = (int)(__brev((unsigned)c) >> 25);  // 7-bit reversal
    S[idx] = (n == rc) ? 1.0f : 0.0f;
  }
  __syncthreads();
  for (int s = 1; s <= 7; ++s) {
    const int bs = 1 << s, h = bs >> 1;
    for (int wi = t; wi < 64 * 128; wi += 256) {
      int p = wi >> 7;       // pair id 0..63
      int c = wi & 127;      // column
      int blk = p >> (s - 1);
      int k = p & (h - 1);
      int n0 = blk * bs + k, n1 = n0 + h;
      float f = 6.283185307179586f * (float)k / (float)bs;
      float cs = cosf(f) + sinf(f);  // cas twiddle
      float e = S[n0 * 128 + c];
      float o = S[n1 * 128 + c];
      float tt = cs * o;
      S[n0 * 128 + c] = e + tt;
      S[n1 * 128 + c] = e - tt;
    }
    __syncthreads();
  }
  for (int idx = t; idx < 128 * 128; idx += 256) T[idx] = S[idx];
}

// ---------------------------------------------------------------------------
// WMMA fp32 16x16x4 fragment layouts (ISA 7.12.2, wave32):
//  A (16x4): lane = half*16+lm ; a.x = A[m][2*half], a.y = A[m][2*half+1]
//  B (4x16): b.x = B[2*half][n], b.y = B[2*half+1][n]
//  C/D (16x16): vgpr r -> row r + 8*half, col = lm
// ---------------------------------------------------------------------------

// Kernel 1: G[img] (64x128) = T[rsel,:] (64x128) * X[img] (128x128)
__global__ void dht_rows(const float* __restrict__ T,
                         const float* __restrict__ x,
                         float* __restrict__ G) {
  const int img  = blockIdx.x;                 // B*Cin images
  const int lane = threadIdx.x & 31;
  const int half = lane >> 4;
  const int lm   = lane & 15;
  const int nc   = threadIdx.x >> 5;           // 8 waves -> 8 col tiles
  const float* X = x + (size_t)img * (MDIM * NDIM);
  float* Gp      = G + (size_t)img * (64 * NDIM);

  for (int mr = 0; mr < 4; ++mr) {             // 4 row tiles of 16
    v8f c = {};
    const int arow = rsel(mr * 16 + lm);       // row of T
    const int col  = nc * 16 + lm;
    for (int k0 = 0; k0 < 128; k0 += 4) {
      v2f a, b;
      a.x = T[arow * 128 + k0 + 2 * half];
      a.y = T[arow * 128 + k0 + 2 * half + 1];
      b.x = X[(k0 + 2 * half) * NDIM + col];
      b.y = X[(k0 + 2 * half + 1) * NDIM + col];
      c = __builtin_amdgcn_wmma_f32_16x16x4_f32(false, a, false, b,
                                                (short)0, c, false, false);
    }
#pragma unroll
    for (int r = 0; r < 8; ++r)
      Gp[(mr * 16 + r + 8 * half) * NDIM + col] = c[r];
  }
}

// Kernel 2: Xc[img] (64x32) = G[img] (64x128) * (T[0:32,:])^T
__global__ void dht_cols(const float* __restrict__ T,
                         const float* __restrict__ G,
                         float* __restrict__ Xc) {
  const int img  = blockIdx.x;
  const int lane = threadIdx.x & 31;
  const int half = lane >> 4;
  const int lm   = lane & 15;
  const int w    = threadIdx.x >> 5;           // 8 waves -> 8 tiles (4x2)
  const int mr   = w >> 1;
  const int nc   = w & 1;
  const float* Gp = G + (size_t)img * (64 * NDIM);
  float* Xp       = Xc + (size_t)img * (64 * 32);

  v8f c = {};
  const int arow = mr * 16 + lm;
  const int col  = nc * 16 + lm;               // output mode column 0..31
  for (int k0 = 0; k0 < 128; k0 += 4) {
    v2f a, b;
    a.x = Gp[arow * NDIM + k0 + 2 * half];
    a.y = Gp[arow * NDIM + k0 + 2 * half + 1];
    // B[k][n] = T[n][k]
    b.x = T[col * 128 + k0 + 2 * half];
    b.y = T[col * 128 + k0 + 2 * half + 1];
    c = __builtin_amdgcn_wmma_f32_16x16x4_f32(false, a, false, b,
                                              (short)0, c, false, false);
  }
#pragma unroll
  for (int r = 0; r < 8; ++r)
    Xp[(mr * 16 + r + 8 * half) * 32 + col] = c[r];
}

// ---------------------------------------------------------------------------
// Kernel 3: mode mixing. out = 0.5*( x1*(w+wn) + x1n*(w-wn) ) summed over Cin.
// Block = (region, o). Weights streamed coalesced (each byte read once);
// the 8-batch x-corner slice for channel i+1 is prefetched into LDS with
// async-to-LDS (ASYNCcnt) while channel i is being accumulated.
// ---------------------------------------------------------------------------
__global__ void spectral(const float* __restrict__ Xc,
                         const float* __restrict__ w1,
                         const float* __restrict__ w2,
                         float* __restrict__ Oc) {
  __shared__ alignas(16) float Xs[2][8 * 1024];  // double buffer, 64 KB
  const int reg = blockIdx.x & 1;
  const int o   = blockIdx.x >> 1;
  const float* W = reg ? w2 : w1;
  const int t = threadIdx.x;                   // 256 threads, 4 modes each

  int mode[4], moden[4];
#pragma unroll
  for (int q = 0; q < 4; ++q) {
    int m = t + 256 * q;
    int xx = m >> 5, yy = m & 31;
    int xn = (32 - xx) & 31, yn = (32 - yy) & 31;
    mode[q]  = m;
    moden[q] = xn * 32 + yn;
  }
  float acc[4][8];
#pragma unroll
  for (int q = 0; q < 4; ++q)
#pragma unroll
    for (int b = 0; b < 8; ++b) acc[q][b] = 0.0f;

  // issue the 32 KB slice Xc[b][i][reg*32 + x][y] (8 batches) into buffer buf
  auto issue = [&](int i, int buf) {
#pragma unroll
    for (int j = 0; j < 8; ++j) {
      int idx = t + 256 * j;                   // 16-byte chunk id, 0..2047
      int b = idx >> 8, pos = idx & 255;       // 256 chunks per batch slice
      const float* gsrc =
          Xc + (size_t)((b * CIN + i) * 64 + reg * 32) * 32 + pos * 4;
      float* ldst = &Xs[buf][idx * 4];
#if HAS_ASYNC_LDS
      __builtin_amdgcn_global_load_async_to_lds_b128(
          (gas_i32x4*)(float*)gsrc, (las_i32x4*)ldst, 0, 0);
#else
      *(float4*)ldst = *(const float4*)gsrc;
#endif
    }
  };
  auto wait_lds = [&]() {
#if HAS_ASYNC_LDS
#if __has_builtin(__builtin_amdgcn_s_wait_asynccnt)
    __builtin_amdgcn_s_wait_asynccnt(0);
#else
    asm volatile("s_wait_asynccnt 0" ::: "memory");
#endif
#endif
    __syncthreads();
  };

  issue(0, 0);
  wait_lds();
  for (int i = 0; i < CIN; ++i) {
    const int cur = i & 1;
    if (i + 1 < CIN) {
      issue(i + 1, cur ^ 1);                   // async prefetch next slice
      __builtin_prefetch(W + (size_t)((i + 1) * COUT + o) * 1024, 0, 0);
    }
    const float* Wi = W + (size_t)(i * COUT + o) * 1024;
    const float* Xb = &Xs[cur][0];
#pragma unroll
    for (int q = 0; q < 4; ++q) {
      float w  = Wi[mode[q]];
      float wn = Wi[moden[q]];
      float wp = w + wn, wm = w - wn;
#pragma unroll
      for (int b = 0; b < 8; ++b)
        acc[q][b] += Xb[b * 1024 + mode[q]] * wp
                   + Xb[b * 1024 + moden[q]] * wm;
    }
    wait_lds();   // drain prefetch + barrier before buffers are reused
  }
#pragma unroll
  for (int q = 0; q < 4; ++q)
#pragma unroll
    for (int b = 0; b < 8; ++b)
      Oc[(size_t)((b * COUT + o) * 64 + reg * 32) * 32 + mode[q]] =
          0.5f * acc[q][b];
}

// ---------------------------------------------------------------------------
// Kernel 4: M1[img] (128x32) = T[:,rsel] (128x64) * Oc[img] (64x32)
// ---------------------------------------------------------------------------
__global__ void final_a(const float* __restrict__ T,
                        const float* __restrict__ Oc,
                        float* __restrict__ M1) {
  const int img  = blockIdx.x;                 // B*Cout images
  const int lane = threadIdx.x & 31;
  const int half = lane >> 4;
  const int lm   = lane & 15;
  const int w    = threadIdx.x >> 5;
  const float* Op = Oc + (size_t)img * (64 * 32);
  float* Mp       = M1 + (size_t)img * (128 * 32);

  for (int tt = 0; tt < 2; ++tt) {             // 16 tiles / 8 waves
    const int tile = w * 2 + tt;
    const int mr = tile >> 1, nc = tile & 1;
    v8f c = {};
    const int arow = mr * 16 + lm;
    const int col  = nc * 16 + lm;
    for (int k0 = 0; k0 < 64; k0 += 4) {
      v2f a, b;
      // A[m][k] = T[m][rsel(k)]
      a.x = T[arow * 128 + rsel(k0 + 2 * half)];
      a.y = T[arow * 128 + rsel(k0 + 2 * half + 1)];
      b.x = Op[(k0 + 2 * half) * 32 + col];
      b.y = Op[(k0 + 2 * half + 1) * 32 + col];
      c = __builtin_amdgcn_wmma_f32_16x16x4_f32(false, a, false, b,
                                                (short)0, c, false, false);
    }
#pragma unroll
    for (int r = 0; r < 8; ++r)
      Mp[(mr * 16 + r + 8 * half) * 32 + col] = c[r];
  }
}

// ---------------------------------------------------------------------------
// Kernel 5: out[img] (128x128) = M1[img] (128x32) * (T[:,0:32])^T / 16384
// ---------------------------------------------------------------------------
__global__ void final_b(const float* __restrict__ T,
                        const float* __restrict__ M1,
                        float* __restrict__ out) {
  const int img  = blockIdx.x;
  const int lane = threadIdx.x & 31;
  const int half = lane >> 4;
  const int lm   = lane & 15;
  const int w    = threadIdx.x >> 5;
  const float* Mp = M1 + (size_t)img * (128 * 32);
  float* Yp       = out + (size_t)img * (MDIM * NDIM);

  for (int tt = 0; tt < 8; ++tt) {             // 64 tiles / 8 waves
    const int tile = w * 8 + tt;
    const int mr = tile >> 3, nc = tile & 7;
    v8f c = {};
    const int arow = mr * 16 + lm;
    const int col  = nc * 16 + lm;
    for (int k0 = 0; k0 < 32; k0 += 4) {
      v2f a, b;
      a.x = Mp[arow * 32 + k0 + 2 * half];
      a.y = Mp[arow * 32 + k0 + 2 * half + 1];
      // B[k][n] = T[n][k]
      b.x = T[col * 128 + k0 + 2 * half];
      b.y = T[col * 128 + k0 + 2 * half + 1];
      c = __builtin_amdgcn_wmma_f32_16x16x4_f32(false, a, false, b,
                                                (short)0, c, false, false);
    }
#pragma unroll
    for (int r = 0; r < 8; ++r)
      Yp[(mr * 16 + r + 8 * half) * NDIM + col] =
          c[r] * (1.0f / (MDIM * NDIM));
  }
}

// ---------------------------------------------------------------------------
extern "C" void kernel_launch(void* const* d_in, const int* in_sizes, int n_in,
                              void* d_out, int out_size, void* d_ws, size_t ws_size,
                              hipStream_t stream) {
  const float* x  = (const float*)d_in[0];   // [8,256,128,128]
  const float* w1 = (const float*)d_in[1];   // [256,256,32,32]
  const float* w2 = (const float*)d_in[2];   // [256,256,32,32]
  float* out = (float*)d_out;                // [8,256,128,128]
  float* ws  = (float*)d_ws;

  // workspace layout (floats)
  float* T  = ws;                               // 16384          (64 KB)
  float* G  = T + 16384;                        // 2048*64*128    (64 MB)
  float* Xc = G + (size_t)2048 * 64 * 128;      // 2048*64*32     (16 MB)
  float* Oc = Xc + (size_t)2048 * 64 * 32;      // 2048*64*32     (16 MB)
  float* M1 = Oc + (size_t)2048 * 64 * 32;      // 2048*128*32    (32 MB)

  build_T <<<1,    256, 0, stream>>>(T);
  dht_rows<<<B_SZ * CIN, 256, 0, stream>>>(T, x, G);
  dht_cols<<<B_SZ * CIN, 256, 0, stream>>>(T, G, Xc);
  spectral<<<2 * COUT,   256, 0, stream>>>(Xc, w1, w2, Oc);
  final_a <<<B_SZ * COUT, 256, 0, stream>>>(T, Oc, M1);
  final_b <<<B_SZ * COUT, 256, 0, stream>>>(T, M1, out);
}